// GGRNNTorch_21629455302666
// MI455X (gfx1250) — compile-verified
//
#include <hip/hip_runtime.h>
#include <hip/hip_bf16.h>

typedef __attribute__((ext_vector_type(2))) float v2f;
typedef __attribute__((ext_vector_type(8))) float v8f;

// ---------------------------------------------------------------------------
// WMMA helpers: V_WMMA_F32_16X16X4_F32, wave32.
// A: 16x4 f32 in 2 VGPRs  (lanes 0-15: M=lane, K={k0,k0+1}; lanes 16-31: K={k0+2,k0+3})
// B: 4x16 f32 in 2 VGPRs  (VGPR v, lanes 0-15: K=v (+2 for lanes 16-31), N=lane&15)
// C/D: 16x16 f32 in 8 VGPRs (VGPR v: M=v (+8 for lanes 16-31), N=lane&15)
// ---------------------------------------------------------------------------
__device__ __forceinline__ v8f wmma4(v2f a, v2f b, v8f c) {
  return __builtin_amdgcn_wmma_f32_16x16x4_f32(false, a, false, b, (short)0, c,
                                               false, false);
}

__device__ __forceinline__ v2f ldA(const float* base, int r0, int k0, int stride,
                                   int lane) {
  int r = r0 + (lane & 15);
  int k = k0 + ((lane >> 4) << 1);
  v2f a;
  a.x = base[r * stride + k];
  a.y = base[r * stride + k + 1];
  return a;
}

__device__ __forceinline__ v2f ldB(const float* base, int k0, int c0, int stride,
                                   int lane) {
  int c = c0 + (lane & 15);
  int k = k0 + ((lane >> 4) << 1);
  v2f b;
  b.x = base[k * stride + c];
  b.y = base[(k + 1) * stride + c];
  return b;
}

__device__ __forceinline__ void stC(float* base, int r0, int c0, int stride,
                                    int lane, v8f c) {
  int r = r0 + ((lane >> 4) << 3);
  int col = c0 + (lane & 15);
#pragma unroll
  for (int v = 0; v < 8; ++v) base[(r + v) * stride + col] = c[v];
}

__device__ __forceinline__ float sigm(float x) {
  return 1.0f / (1.0f + __expf(-x));
}

// ---------------------------------------------------------------------------
// Gate GEMM for the graph-gated GRU.
// sG layout [64 x 512]: cols 0-127   = gx_z + gh_z
//                       cols 128-255 = gx_r + gh_r
//                       cols 256-383 = gx_n (+ bias)
//                       cols 384-511 = gh_n            (n-gate needs r*gh_n!)
// 32 column-jobs: j<16 summed (m@W + h@U + b), j 16-23 gx_n (m@W + b),
// j 24-31 gh_n (h@U only).
// ---------------------------------------------------------------------------
__device__ __forceinline__ void gate_gemm(const float* sMm, int mStride, int Km,
                                          const float* sHh,
                                          const float* __restrict__ Wg,
                                          const float* __restrict__ Ug,
                                          const float* __restrict__ bg,
                                          float* sGg, int wave, int lane) {
  for (int j = wave; j < 32; j += 8) {
    const bool doW = (j < 24);
    const bool doU = (j < 16) || (j >= 24);
    const int cW = j * 16;
    const int cU = (j < 16) ? j * 16 : 256 + (j - 24) * 16;
    const int cO = (j < 24) ? j * 16 : 384 + (j - 24) * 16;
    float bv = doW ? bg[cW + (lane & 15)] : 0.0f;
    v8f a0, a1, a2, a3;
#pragma unroll
    for (int v = 0; v < 8; ++v) { a0[v] = bv; a1[v] = bv; a2[v] = bv; a3[v] = bv; }
    if (doW) {
      for (int k = 0; k < Km; k += 4) {
        v2f bf = ldB(Wg, k, cW, 384, lane);
        a0 = wmma4(ldA(sMm, 0, k, mStride, lane), bf, a0);
        a1 = wmma4(ldA(sMm, 16, k, mStride, lane), bf, a1);
        a2 = wmma4(ldA(sMm, 32, k, mStride, lane), bf, a2);
        a3 = wmma4(ldA(sMm, 48, k, mStride, lane), bf, a3);
      }
    }
    if (doU) {
      for (int k = 0; k < 128; k += 4) {
        v2f bf = ldB(Ug, k, cU, 384, lane);
        a0 = wmma4(ldA(sHh, 0, k, 128, lane), bf, a0);
        a1 = wmma4(ldA(sHh, 16, k, 128, lane), bf, a1);
        a2 = wmma4(ldA(sHh, 32, k, 128, lane), bf, a2);
        a3 = wmma4(ldA(sHh, 48, k, 128, lane), bf, a3);
      }
    }
    stC(sGg, 0, cO, 512, lane, a0);
    stC(sGg, 16, cO, 512, lane, a1);
    stC(sGg, 32, cO, 512, lane, a2);
    stC(sGg, 48, cO, 512, lane, a3);
  }
}

__device__ __forceinline__ void gru_update(const float* sGg, float* sHh, int tid) {
  for (int i = tid; i < 64 * 128; i += 256) {
    int r = i >> 7, c = i & 127;
    const float* g = sGg + r * 512;
    float z = sigm(g[c]);
    float rr = sigm(g[128 + c]);
    float nn = tanhf(g[256 + c] + rr * g[384 + c]);
    sHh[i] = (1.0f - z) * nn + z * sHh[i];
  }
}

// ---------------------------------------------------------------------------
// Fused GGRNN layer1 + layer2 + node-maxpool. One block per batch element;
// h1/h2 persist in LDS across all 64 time steps. ~256KB LDS of the 320KB/WG.
// ---------------------------------------------------------------------------
__global__ __launch_bounds__(256, 1) void ggrnn_fused_kernel(
    const float* __restrict__ X,   // [64,48,64,64]
    const float* __restrict__ A,   // [64,64,64]
    const float* __restrict__ W1, const float* __restrict__ U1,
    const float* __restrict__ b1, const float* __restrict__ W2,
    const float* __restrict__ U2, const float* __restrict__ b2,
    float* __restrict__ seq)       // [64,64,256], writes cols 0-127
{
  __shared__ float sA[64 * 64];     // 16 KB  adjacency (resident all steps)
  __shared__ float sH1[64 * 128];   // 32 KB  layer-1 hidden
  __shared__ float sH2[64 * 128];   // 32 KB  layer-2 hidden
  __shared__ float sX[64 * 64];     // 16 KB  x_t
  __shared__ float sM[64 * 128];    // 32 KB  A@x scratch
  __shared__ float sG[64 * 512];    // 128 KB gate buffer

  const int b = blockIdx.x;
  const int tid = threadIdx.x, wave = tid >> 5, lane = tid & 31;

  const float* Ab = A + (size_t)b * 4096;
  for (int i = tid; i < 4096; i += 256) sA[i] = Ab[i];
  for (int i = tid; i < 8192; i += 256) { sH1[i] = 0.f; sH2[i] = 0.f; }
  __syncthreads();

  for (int t = 0; t < 64; ++t) {
    // time is front-padded: t<16 -> zeros, else concatenation[b, t-16]
    if (t < 16) {
      for (int i = tid; i < 4096; i += 256) sX[i] = 0.f;
    } else {
      const float* xp = X + (size_t)(b * 48 + (t - 16)) * 4096;
      for (int i = tid; i < 4096; i += 256) sX[i] = xp[i];
    }
    __syncthreads();

    // ---- layer 1: m = A @ x_t  (64x64, K=64) ----
    for (int tt = wave; tt < 16; tt += 8) {
      int r0 = (tt >> 2) * 16, c0 = (tt & 3) * 16;
      v8f acc = {};
      for (int k = 0; k < 64; k += 4)
        acc = wmma4(ldA(sA, r0, k, 64, lane), ldB(sX, k, c0, 64, lane), acc);
      stC(sM, r0, c0, 64, lane, acc);
    }
    __syncthreads();
    gate_gemm(sM, 64, 64, sH1, W1, U1, b1, sG, wave, lane);
    __syncthreads();
    gru_update(sG, sH1, tid);
    __syncthreads();

    // ---- layer 2: m = A @ h1  (64x128, K=64) ----
    for (int tt = wave; tt < 32; tt += 8) {
      int r0 = (tt >> 3) * 16, c0 = (tt & 7) * 16;
      v8f acc = {};
      for (int k = 0; k < 64; k += 4)
        acc = wmma4(ldA(sA, r0, k, 64, lane), ldB(sH1, k, c0, 128, lane), acc);
      stC(sM, r0, c0, 128, lane, acc);
    }
    __syncthreads();
    gate_gemm(sM, 128, 128, sH2, W2, U2, b2, sG, wave, lane);
    __syncthreads();
    gru_update(sG, sH2, tid);
    __syncthreads();

    // ---- max pool over nodes -> seq[b,t,0:128] ----
    if (tid < 128) {
      float mx = -3.4e38f;
      for (int r = 0; r < 64; ++r) mx = fmaxf(mx, sH2[r * 128 + tid]);
      seq[(size_t)(b * 64 + t) * 256 + tid] = mx;
    }
    __syncthreads();
  }
}

// ---------------------------------------------------------------------------
// Embedding gather into seq cols 128-255 (front pad -> index 0 rows).
// ---------------------------------------------------------------------------
__global__ __launch_bounds__(256) void emb_kernel(
    const int* __restrict__ attrs,  // [64,48,4]
    const float* __restrict__ E0, const float* __restrict__ E1,
    const float* __restrict__ E2, const float* __restrict__ E3,
    float* __restrict__ seq) {
  int i = blockIdx.x * blockDim.x + threadIdx.x;  // over 64*64*128
  if (i >= 64 * 64 * 128) return;
  int c = i & 127;
  int t = (i >> 7) & 63;
  int b = i >> 13;
  int e = c >> 5, off = c & 31;
  int idx = (t < 16) ? 0 : attrs[(size_t)(b * 48 + (t - 16)) * 4 + e];
  const float* E = (e == 0) ? E0 : (e == 1) ? E1 : (e == 2) ? E2 : E3;
  seq[(size_t)(b * 64 + t) * 256 + 128 + c] = E[idx * 32 + off];
}

// ---------------------------------------------------------------------------
// LSTM head: one block (16 waves) carries h/c for all 64 batch rows in LDS.
// Per step: g = x@Wx + h@Wh + b  ([64x384]@[384x512] via WMMA f32).
// ---------------------------------------------------------------------------
__global__ __launch_bounds__(512, 1) void lstm_head_kernel(
    const float* __restrict__ seq,  // [64,64,256]
    const float* __restrict__ Wx,   // [256,512]
    const float* __restrict__ Wh,   // [128,512]
    const float* __restrict__ bl,   // [512]
    const float* __restrict__ Wo,   // [128,33]
    const float* __restrict__ bo,   // [33]
    float* __restrict__ out)        // [64,33]
{
  __shared__ float sX[64 * 256];    // 64 KB
  __shared__ float sH[64 * 128];    // 32 KB
  __shared__ float sC[64 * 128];    // 32 KB
  __shared__ float sG[64 * 512];    // 128 KB
  const int tid = threadIdx.x, wave = tid >> 5, lane = tid & 31;

  for (int i = tid; i < 8192; i += 512) { sH[i] = 0.f; sC[i] = 0.f; }
  __syncthreads();

  for (int t = 0; t < 64; ++t) {
    for (int i = tid; i < 64 * 256; i += 512) {
      int r = i >> 8, c = i & 255;
      sX[i] = seq[(size_t)(r * 64 + t) * 256 + c];
    }
    __syncthreads();

    for (int j = wave; j < 32; j += 16) {
      int c0 = j * 16;
      float bv = bl[c0 + (lane & 15)];
      v8f a0, a1, a2, a3;
#pragma unroll
      for (int v = 0; v < 8; ++v) { a0[v] = bv; a1[v] = bv; a2[v] = bv; a3[v] = bv; }
      for (int k = 0; k < 256; k += 4) {
        v2f bf = ldB(Wx, k, c0, 512, lane);
        a0 = wmma4(ldA(sX, 0, k, 256, lane), bf, a0);
        a1 = wmma4(ldA(sX, 16, k, 256, lane), bf, a1);
        a2 = wmma4(ldA(sX, 32, k, 256, lane), bf, a2);
        a3 = wmma4(ldA(sX, 48, k, 256, lane), bf, a3);
      }
      for (int k = 0; k < 128; k += 4) {
        v2f bf = ldB(Wh, k, c0, 512, lane);
        a0 = wmma4(ldA(sH, 0, k, 128, lane), bf, a0);
        a1 = wmma4(ldA(sH, 16, k, 128, lane), bf, a1);
        a2 = wmma4(ldA(sH, 32, k, 128, lane), bf, a2);
        a3 = wmma4(ldA(sH, 48, k, 128, lane), bf, a3);
      }
      stC(sG, 0, c0, 512, lane, a0);
      stC(sG, 16, c0, 512, lane, a1);
      stC(sG, 32, c0, 512, lane, a2);
      stC(sG, 48, c0, 512, lane, a3);
    }
    __syncthreads();

    for (int i = tid; i < 8192; i += 512) {
      int r = i >> 7, c = i & 127;
      const float* g = sG + r * 512;
      float ig = sigm(g[c]);
      float fg = sigm(g[128 + c]);
      float gg = tanhf(g[256 + c]);
      float og = sigm(g[384 + c]);
      float cc = fg * sC[i] + ig * gg;
      sC[i] = cc;
      sH[i] = og * tanhf(cc);
    }
    __syncthreads();
  }

  // classifier: [64,128] @ [128,33] + bo  (tiny -> scalar dot products)
  for (int i = tid; i < 64 * 33; i += 512) {
    int r = i / 33, c = i % 33;
    float s = bo[c];
    for (int k = 0; k < 128; ++k) s += sH[r * 128 + k] * Wo[k * 33 + c];
    out[i] = s;
  }
}

// ---------------------------------------------------------------------------
extern "C" void kernel_launch(void* const* d_in, const int* in_sizes, int n_in,
                              void* d_out, int out_size, void* d_ws,
                              size_t ws_size, hipStream_t stream) {
  const float* X = (const float*)d_in[0];     // concatenation [64,48,64,64]
  const float* A = (const float*)d_in[1];     // A_in [64,64,64]
  // d_in[2] last_places_activated, d_in[3] multiple_outputs: unused
  const int* attrs = (const int*)d_in[4];     // X_attributes [64,48,4]
  const float* W1 = (const float*)d_in[5];
  const float* U1 = (const float*)d_in[6];
  const float* b1 = (const float*)d_in[7];
  const float* W2 = (const float*)d_in[8];
  const float* U2 = (const float*)d_in[9];
  const float* b2 = (const float*)d_in[10];
  const float* E0 = (const float*)d_in[11];
  const float* E1 = (const float*)d_in[12];
  const float* E2 = (const float*)d_in[13];
  const float* E3 = (const float*)d_in[14];
  const float* Wx = (const float*)d_in[15];
  const float* Wh = (const float*)d_in[16];
  const float* bl = (const float*)d_in[17];
  const float* Wo = (const float*)d_in[18];
  const float* bo = (const float*)d_in[19];

  float* seq = (float*)d_ws;  // [64,64,256] fp32 = 4 MB scratch
  float* out = (float*)d_out; // [64,33] fp32

  ggrnn_fused_kernel<<<64, 256, 0, stream>>>(X, A, W1, U1, b1, W2, U2, b2, seq);
  emb_kernel<<<(64 * 64 * 128 + 255) / 256, 256, 0, stream>>>(attrs, E0, E1, E2,
                                                              E3, seq);
  lstm_head_kernel<<<1, 512, 0, stream>>>(seq, Wx, Wh, bl, Wo, bo, out);
}